// SemiConv2d_15341623181422
// MI455X (gfx1250) — compile-verified
//
#include <hip/hip_runtime.h>

// Tropical (max-min) 3x3 "same" conv, N=8, C=O=32, H=W=112, fp32.
// VALU-issue-bound (925M max-min candidates; memory ~1us at 23.3TB/s).
// Design: LDS-tiled input (async global->LDS staging), weights transposed
// so each tap's 32 weights are wave-uniform SMEM scalars, inner loop =
// one fused v_maxmin_num_f32 per candidate (probe-confirmed on gfx1250).
// 784 blocks to cover all WGPs on a large CDNA5 part.

#define KS      3
#define CIN     32
#define COUT    32
#define HH      112
#define WW      112
#define NB      8
#define TILE_W  16
#define TILE_H  8
#define HALO_W  (TILE_W + KS - 1)        // 18
#define HALO_H  (TILE_H + KS - 1)        // 10
#define CH_STRIDE (HALO_H * HALO_W)      // 180 floats per channel tile
#define NTHREADS (TILE_W * TILE_H)       // 128 = 4 waves
#define NWT     (COUT * CIN * KS * KS)   // 9216 weights

#ifndef NEG_INF_F
#define NEG_INF_F (-__builtin_inff())
#endif

// gfx1250 has the fused max(min(a,b),c) op (probe-confirmed round 3).
#define TRY_MAXMIN_ASM 1

// ---- CDNA5 async global->LDS staging (guarded; sync fallback) ----
#if defined(__has_builtin)
#  if __has_builtin(__builtin_amdgcn_global_load_async_to_lds_b32) && \
      __has_builtin(__builtin_amdgcn_s_wait_asynccnt)
#    define USE_ASYNC_LDS 1
#  endif
#endif
#ifndef USE_ASYNC_LDS
#  define USE_ASYNC_LDS 0
#endif

typedef __attribute__((address_space(1))) int* gptr_i32;
typedef __attribute__((address_space(3))) int* lptr_i32;

__device__ __forceinline__ void maxmin_acc(float& acc, float xv, float wv) {
#if TRY_MAXMIN_ASM
    // acc = max(min(xv, wv), acc); VOP3, weight rides the scalar source.
    asm("v_maxmin_num_f32 %0, %1, %2, %0" : "+v"(acc) : "v"(xv), "s"(wv));
#else
    acc = fmaxf(acc, fminf(xv, wv));
#endif
}

// Transpose weights [o][c][kh][kw] -> [c*9+kh*3+kw][o] so the 32 weights of a
// tap are contiguous + wave-uniform (SMEM s_load candidates).
__global__ void SemiConv2d_wt_kernel(const float* __restrict__ w,
                                     float* __restrict__ wT) {
    int t = blockIdx.x * blockDim.x + threadIdx.x;
    if (t >= NWT) return;
    int o  = t & (COUT - 1);
    int ck = t >> 5;                       // c*9 + kh*3 + kw
    wT[ck * COUT + o] = w[o * (CIN * KS * KS) + ck];
}

template <bool TRANSPOSED>
__global__ __launch_bounds__(NTHREADS)
void SemiConv2d_main_kernel(const float* __restrict__ x,
                            const float* __restrict__ wq,   // wT or original w
                            float* __restrict__ out) {
    __shared__ float tile[CIN * CH_STRIDE];   // 23040 B

    const int tx0 = blockIdx.x * TILE_W;
    const int ty0 = blockIdx.y * TILE_H;
    const int n   = blockIdx.z;
    const int tid = threadIdx.x;

    // ---- Stage 32-channel 10x18 halo tile into LDS, -inf padding ----
    const float* xn = x + (size_t)n * CIN * HH * WW;
    for (int idx = tid; idx < CIN * CH_STRIDE; idx += NTHREADS) {
        int c  = idx / CH_STRIDE;
        int r  = idx - c * CH_STRIDE;
        int iy = r / HALO_W;
        int ix = r - iy * HALO_W;
        int gy = ty0 + iy - 1;
        int gx = tx0 + ix - 1;
        bool ok = (unsigned)gy < (unsigned)HH && (unsigned)gx < (unsigned)WW;
#if USE_ASYNC_LDS
        if (ok) {
            __builtin_amdgcn_global_load_async_to_lds_b32(
                (gptr_i32)(xn + ((size_t)c * HH + gy) * WW + gx),
                (lptr_i32)&tile[idx],
                0, 0);
        } else {
            tile[idx] = NEG_INF_F;
        }
#else
        tile[idx] = ok ? xn[((size_t)c * HH + gy) * WW + gx] : NEG_INF_F;
#endif
    }
#if USE_ASYNC_LDS
    __builtin_amdgcn_s_wait_asynccnt(0);
#endif
    __syncthreads();

    const int tx = tid & (TILE_W - 1);
    const int ty = tid >> 4;

    float acc[COUT];
#pragma unroll
    for (int o = 0; o < COUT; ++o) acc[o] = NEG_INF_F;

    // ---- Main loop: per tap, 1 ds_load + 32 fused v_maxmin_num_f32 ----
    for (int c = 0; c < CIN; ++c) {
        const float* tc = &tile[c * CH_STRIDE];
#pragma unroll
        for (int kh = 0; kh < KS; ++kh) {
#pragma unroll
            for (int kw = 0; kw < KS; ++kw) {
                const float xv = tc[(ty + kh) * HALO_W + tx + kw];
                const int ck = (c * KS + kh) * KS + kw;
#pragma unroll
                for (int o = 0; o < COUT; ++o) {
                    const float wv = TRANSPOSED ? wq[ck * COUT + o]
                                                : wq[o * (CIN * KS * KS) + ck];
                    maxmin_acc(acc[o], xv, wv);
                }
            }
        }
    }

    // ---- Write 32 output channels (coalesced across lanes) ----
    float* ob = out + (((size_t)n * COUT) * HH + (ty0 + ty)) * WW + (tx0 + tx);
#pragma unroll
    for (int o = 0; o < COUT; ++o) {
        ob[(size_t)o * HH * WW] = acc[o];
    }
}

extern "C" void kernel_launch(void* const* d_in, const int* in_sizes, int n_in,
                              void* d_out, int out_size, void* d_ws, size_t ws_size,
                              hipStream_t stream) {
    const float* x = (const float*)d_in[0];
    const float* w = (const float*)d_in[1];
    float* out = (float*)d_out;

    const dim3 grid(WW / TILE_W, HH / TILE_H, NB);   // 7 x 14 x 8 = 784 blocks
    const dim3 block(NTHREADS);                       // 128 = 4 waves

    if (ws_size >= (size_t)NWT * sizeof(float)) {
        float* wT = (float*)d_ws;
        SemiConv2d_wt_kernel<<<dim3((NWT + 255) / 256), dim3(256), 0, stream>>>(w, wT);
        SemiConv2d_main_kernel<true><<<grid, block, 0, stream>>>(x, wT, out);
    } else {
        SemiConv2d_main_kernel<false><<<grid, block, 0, stream>>>(x, w, out);
    }
}